// GNNEncoder_light_31284541784162
// MI455X (gfx1250) — compile-verified
//
#include <hip/hip_runtime.h>

typedef __attribute__((ext_vector_type(2))) float v2f;
typedef __attribute__((ext_vector_type(8))) float v8f;
typedef __attribute__((ext_vector_type(4))) unsigned u32x4;
typedef __attribute__((ext_vector_type(8))) unsigned u32x8;

#define WMMA_F32(a, b, c) \
  __builtin_amdgcn_wmma_f32_16x16x4_f32(false, (a), false, (b), (short)0, (c), false, false)

constexpr int Bsz = 2, SC = 512, ST = 512, H = 64;
constexpr int LP = 68;              // padded LDS row stride (floats): 64 + 4 -> bank-conflict-free
constexpr float EPS = 1e-5f;

// workspace layout (float offsets)
constexpr int OFF_UH1 = 0;
constexpr int OFF_UH2 = 65536;
constexpr int OFF_VH1 = 131072;
constexpr int OFF_VH2 = 196608;
constexpr int OFF_AH1 = 262144;
constexpr int OFF_BH2 = 327680;
constexpr int OFF_H1ACC = 393216;
constexpr int OFF_H2ACC = 458752;
constexpr int OFF_ESUM = 524288;   // 64
constexpr int OFF_ESQ  = 524352;   // 64
constexpr int OFF_H1M  = 524416;
constexpr int OFF_H1R  = 524480;
constexpr int OFF_H2M  = 524544;
constexpr int OFF_H2R  = 524608;
constexpr int OFF_EM   = 524672;
constexpr int OFF_ER   = 524736;
constexpr int ZERO_COUNT = 131200;  // [OFF_H1ACC, OFF_ESQ+64)

// ---- CDNA5 data movers ------------------------------------------------------

// TDM: DMA W_C (64x64 f32, contiguous) into LDS with row padding 64->68 DWORDs.
// D# per CDNA5 ISA: group0 {count,lds,global,type}, group1
// {data_size=4B, pad_enable, pad_interval=64dw, pad_amount=4dw, dims}.
__device__ __forceinline__ void tdm_load_w_padded(const float* W, float* sW) {
  unsigned long long ga = (unsigned long long)(uintptr_t)W;
  unsigned lds = (unsigned)(uintptr_t)sW;
  u32x4 g0;
  g0[0] = 1u;                                              // count=1, user mode
  g0[1] = lds;                                             // lds_addr
  g0[2] = (unsigned)ga;                                    // global_addr[31:0]
  g0[3] = (unsigned)((ga >> 32) & 0x01FFFFFFull) | 0x80000000u;  // addr[56:32] | type=2
  u32x8 g1;
  g1[0] = (2u << 16)            // data_size = 4 bytes
        | (1u << 20)            // pad_enable
        | (5u << 22)            // pad_interval: 64 DWORDs
        | (3u << 25);           // pad_amount: 4 DWORDs
  g1[1] = (4096u & 0xFFFFu) << 16;        // tensor_dim0[15:0] = 4096
  g1[2] = (4096u >> 16) | (1u << 16);     // tensor_dim0[31:16] | tensor_dim1 = 1
  g1[3] = (4096u << 16);                  // tensor_dim1 hi | tile_dim0 = 4096
  g1[4] = 1u;                             // tile_dim1 = 1, tile_dim2 = 0
  g1[5] = 4096u;                          // tensor_dim0_stride[31:0]
  g1[6] = 0u;
  g1[7] = 0u;
  asm volatile("tensor_load_to_lds %0, %1" :: "s"(g0), "s"(g1) : "memory");
}

// Async-copy one e chunk (16 rows x 64 f32, contiguous in global) into LDS with
// padded rows (stride LP). 8 x b128: each copies 2 rows (512B), fully coalesced.
__device__ __forceinline__ void async_tile(unsigned ldsbase, const float* g, int lane) {
  int r2 = lane >> 4;             // row within pair
  int col = (lane & 15) * 4;      // float column (16B granules)
#pragma unroll
  for (int j = 0; j < 8; ++j) {
    int row = j * 2 + r2;
    asm volatile("global_load_async_to_lds_b128 %0, %1, off"
                 :: "v"(ldsbase + (unsigned)((row * LP + col) * 4)),
                    "v"(g + (long)row * H + col)
                 : "memory");
  }
}

__device__ __forceinline__ void wait_async_le8() {
  asm volatile("s_wait_asynccnt 0x8" ::: "memory");
}
__device__ __forceinline__ void wait_async_0() {
  asm volatile("s_wait_asynccnt 0x0" ::: "memory");
}

// -----------------------------------------------------------------------------

__global__ __launch_bounds__(256) void zero_acc(float* ws) {
  int i = blockIdx.x * 256 + threadIdx.x;
  if (i < ZERO_COUNT) ws[OFF_H1ACC + i] = 0.0f;
}

struct GemmArgs {
  const float* x[6];
  const float* W[6];
  const float* bias[6];
  float* out[6];
};

// Six [1024,64] x [64,64]^T GEMMs: one wave per 16-row M tile, N=64 (4 accums)
__global__ __launch_bounds__(256) void small_gemms(GemmArgs ga) {
  int lane = threadIdx.x & 31;
  int wv = threadIdx.x >> 5;
  int task = blockIdx.x * 8 + wv;   // 0..383
  int mat = task >> 6;              // 0..5
  int m0 = (task & 63) << 4;
  const float* x = ga.x[mat];
  const float* W = ga.W[mat];
  const float* bias = ga.bias[mat];
  float* out = ga.out[mat];
  int half = lane >> 4, lr = lane & 15;
  const float* arow = x + (m0 + lr) * H + half * 2;
  v8f acc[4] = {};
  for (int kk = 0; kk < 16; ++kk) {
    v2f a = *(const v2f*)(arow + 4 * kk);
#pragma unroll
    for (int nt = 0; nt < 4; ++nt) {
      v2f bb = *(const v2f*)(W + (nt * 16 + lr) * H + 4 * kk + half * 2);
      acc[nt] = WMMA_F32(a, bb, acc[nt]);
    }
  }
#pragma unroll
  for (int nt = 0; nt < 4; ++nt) {
    int n = nt * 16 + lr;
    float bv = bias[n];
#pragma unroll
    for (int v = 0; v < 8; ++v) {
      int row = m0 + v + 8 * half;
      out[row * H + n] = acc[nt][v] + bv;
    }
  }
}

// Pass 1: block = 4 waves; wave owns one c-row, loops 8 s-chunks of 16.
// e tiles double-buffered in LDS via async loads; W_C in LDS via TDM (padded).
__global__ __launch_bounds__(128) void pass1(const float* __restrict__ e,
                                             const float* __restrict__ WC,
                                             const float* __restrict__ bC,
                                             float* __restrict__ ws) {
  __shared__ __align__(128) float sW[64 * LP];
  __shared__ __align__(128) float sE[4][2][16 * LP];
  __shared__ float sH2[16 * 64];
  __shared__ float sStat[128];
  int tid = threadIdx.x, lane = tid & 31, wv = tid >> 5;
  int bid = blockIdx.x;
  int sb = bid & 3;
  int cb = (bid >> 2) & 127;
  int b = bid >> 9;
  int c = cb * 4 + wv;
  int half = lane >> 4, lr = lane & 15;

  if (wv == 0) {
    tdm_load_w_padded(WC, sW);
    __builtin_amdgcn_s_wait_tensorcnt(0);
  }
  sStat[tid] = 0.0f;

  const float* Ah1 = ws + OFF_AH1;
  const float* Bh2 = ws + OFF_BH2;
  const float* Vh1 = ws + OFF_VH1;
  const float* Vh2 = ws + OFF_VH2;

  long rowc = (long)(b * SC + c);
  const float* etile = e + rowc * (long)ST * H;

  unsigned eb[2] = { (unsigned)(uintptr_t)&sE[wv][0][0],
                     (unsigned)(uintptr_t)&sE[wv][1][0] };
  async_tile(eb[0], etile + (long)(sb * 128) * H, lane);

  float badd[4], vh1r[4];
  float h1sum[4] = {0.f, 0.f, 0.f, 0.f};
  float esum[4] = {0.f, 0.f, 0.f, 0.f};
  float esq[4] = {0.f, 0.f, 0.f, 0.f};
#pragma unroll
  for (int nt = 0; nt < 4; ++nt) {
    int n = nt * 16 + lr;
    badd[nt] = bC[n] + Ah1[rowc * H + n];
    vh1r[nt] = Vh1[rowc * H + n];
  }
  __syncthreads();  // sW (TDM) + sStat visible

  for (int iter = 0; iter < 8; ++iter) {
    int s0 = sb * 128 + iter * 16;
    if (iter < 7) async_tile(eb[(iter + 1) & 1], etile + (long)(s0 + 16) * H, lane);
#pragma unroll
    for (int i = 0; i < 8; ++i) sH2[tid * 8 + i] = 0.0f;
    __syncthreads();

    if (iter < 7) wait_async_le8(); else wait_async_0();
    const float* sEb = &sE[wv][iter & 1][0];
    v8f acc[4] = {};
    for (int kk = 0; kk < 16; ++kk) {
      v2f a = *(const v2f*)(sEb + lr * LP + 4 * kk + half * 2);
#pragma unroll
      for (int nt = 0; nt < 4; ++nt) {
        v2f bb = *(const v2f*)(sW + (nt * 16 + lr) * LP + 4 * kk + half * 2);
        acc[nt] = WMMA_F32(a, bb, acc[nt]);
      }
    }
#pragma unroll
    for (int nt = 0; nt < 4; ++nt) {
      int n = nt * 16 + lr;
#pragma unroll
      for (int v = 0; v < 8; ++v) {
        int sl = v + 8 * half;  // 0..15
        int s = s0 + sl;
        long rs = (long)(b * ST + s) * H + n;
        float x = acc[nt][v] + badd[nt] + Bh2[rs];
        esum[nt] += x;
        esq[nt] += x * x;
        float g = 1.0f / (1.0f + __expf(-x));
        h1sum[nt] += g * Vh2[rs];
        atomicAdd(&sH2[sl * H + n], g * vh1r[nt]);
      }
    }
    __syncthreads();
#pragma unroll
    for (int i = 0; i < 8; ++i) {
      int idx = tid * 8 + i;
      int sl = idx >> 6, n = idx & 63;
      atomicAdd(ws + OFF_H2ACC + (long)(b * ST + s0 + sl) * H + n, sH2[idx]);
    }
    __syncthreads();
  }

#pragma unroll
  for (int nt = 0; nt < 4; ++nt) {
    float hs = h1sum[nt] + __shfl_xor(h1sum[nt], 16, 32);
    float es = esum[nt] + __shfl_xor(esum[nt], 16, 32);
    float eq = esq[nt] + __shfl_xor(esq[nt], 16, 32);
    if (half == 0) {
      int n = nt * 16 + lr;
      atomicAdd(ws + OFF_H1ACC + rowc * H + n, hs);
      atomicAdd(&sStat[n], es);
      atomicAdd(&sStat[64 + n], eq);
    }
  }
  __syncthreads();
  atomicAdd(ws + OFF_ESUM + tid, sStat[tid]);
}

// BN statistics: block 0 -> h1n, block 1 -> h2n, block 2 -> e (from atomics)
__global__ __launch_bounds__(256) void stats(float* ws) {
  __shared__ float sS[256], sQ[256];
  int t = threadIdx.x;
  int blk = blockIdx.x;
  if (blk < 2) {
    const float* U = ws + (blk == 0 ? OFF_UH1 : OFF_UH2);
    const float* A = ws + (blk == 0 ? OFF_H1ACC : OFF_H2ACC);
    int n = t & 63, r0 = t >> 6;
    float s = 0.f, q = 0.f;
    for (int r = r0; r < 1024; r += 4) {
      float x = U[r * 64 + n] + A[r * 64 + n];
      s += x;
      q += x * x;
    }
    sS[t] = s;
    sQ[t] = q;
    __syncthreads();
    if (t < 64) {
      float S = sS[t] + sS[t + 64] + sS[t + 128] + sS[t + 192];
      float Q = sQ[t] + sQ[t + 64] + sQ[t + 128] + sQ[t + 192];
      float m = S * (1.0f / 1024.0f);
      float v = Q * (1.0f / 1024.0f) - m * m;
      ws[(blk == 0 ? OFF_H1M : OFF_H2M) + t] = m;
      ws[(blk == 0 ? OFF_H1R : OFF_H2R) + t] = rsqrtf(v + EPS);
    }
  } else {
    if (t < 64) {
      const float invN = 1.0f / (float)(Bsz * SC * ST);
      float m = ws[OFF_ESUM + t] * invN;
      float v = ws[OFF_ESQ + t] * invN - m * m;
      ws[OFF_EM + t] = m;
      ws[OFF_ER + t] = rsqrtf(v + EPS);
    }
  }
}

__global__ __launch_bounds__(256) void apply_h(const float* __restrict__ h1,
                                               const float* __restrict__ h2,
                                               const float* __restrict__ gamma,
                                               const float* __restrict__ beta,
                                               const float* __restrict__ ws,
                                               float* __restrict__ out) {
  int idx = blockIdx.x * 256 + threadIdx.x;
  if (idx >= 131072) return;
  int which = idx >> 16;
  int i = idx & 65535;
  int n = i & 63;
  const float* U = ws + (which ? OFF_UH2 : OFF_UH1);
  const float* A = ws + (which ? OFF_H2ACC : OFF_H1ACC);
  float m = ws[(which ? OFF_H2M : OFF_H1M) + n];
  float r = ws[(which ? OFF_H2R : OFF_H1R) + n];
  float x = U[i] + A[i];
  float y = (x - m) * r * gamma[n] + beta[n];
  float h = which ? h2[i] : h1[i];
  out[idx] = h + fmaxf(y, 0.0f);
}

// Pass 2: recompute e_new tiles via WMMA (A + residual e both from the async-
// staged LDS tile), apply BN+ReLU+residual, write eo.
__global__ __launch_bounds__(128) void pass2(const float* __restrict__ e,
                                             const float* __restrict__ WC,
                                             const float* __restrict__ bC,
                                             const float* __restrict__ gamma_e,
                                             const float* __restrict__ beta_e,
                                             const float* __restrict__ ws,
                                             float* __restrict__ out) {
  __shared__ __align__(128) float sW[64 * LP];
  __shared__ __align__(128) float sE[4][2][16 * LP];
  int tid = threadIdx.x, lane = tid & 31, wv = tid >> 5;
  int bid = blockIdx.x;
  int sb = bid & 3;
  int cb = (bid >> 2) & 127;
  int b = bid >> 9;
  int c = cb * 4 + wv;
  int half = lane >> 4, lr = lane & 15;

  if (wv == 0) {
    tdm_load_w_padded(WC, sW);
    __builtin_amdgcn_s_wait_tensorcnt(0);
  }

  long rowc = (long)(b * SC + c);
  const float* etile = e + rowc * (long)ST * H;
  unsigned eb[2] = { (unsigned)(uintptr_t)&sE[wv][0][0],
                     (unsigned)(uintptr_t)&sE[wv][1][0] };
  async_tile(eb[0], etile + (long)(sb * 128) * H, lane);

  const float* Ah1 = ws + OFF_AH1;
  const float* Bh2 = ws + OFF_BH2;
  float badd[4], t1[4], t0[4];
#pragma unroll
  for (int nt = 0; nt < 4; ++nt) {
    int n = nt * 16 + lr;
    badd[nt] = bC[n] + Ah1[rowc * H + n];
    float rr = ws[OFF_ER + n] * gamma_e[n];
    t1[nt] = rr;
    t0[nt] = beta_e[n] - ws[OFF_EM + n] * rr;
  }
  __syncthreads();  // sW visible

  for (int iter = 0; iter < 8; ++iter) {
    int s0 = sb * 128 + iter * 16;
    if (iter < 7) async_tile(eb[(iter + 1) & 1], etile + (long)(s0 + 16) * H, lane);
    if (iter < 7) wait_async_le8(); else wait_async_0();
    const float* sEb = &sE[wv][iter & 1][0];
    v8f acc[4] = {};
    for (int kk = 0; kk < 16; ++kk) {
      v2f a = *(const v2f*)(sEb + lr * LP + 4 * kk + half * 2);
#pragma unroll
      for (int nt = 0; nt < 4; ++nt) {
        v2f bb = *(const v2f*)(sW + (nt * 16 + lr) * LP + 4 * kk + half * 2);
        acc[nt] = WMMA_F32(a, bb, acc[nt]);
      }
    }
#pragma unroll
    for (int nt = 0; nt < 4; ++nt) {
      int n = nt * 16 + lr;
#pragma unroll
      for (int v = 0; v < 8; ++v) {
        int sl = v + 8 * half;
        int s = s0 + sl;
        long off = (rowc * ST + s) * (long)H + n;
        float x = acc[nt][v] + badd[nt] + Bh2[(long)(b * ST + s) * H + n];
        float y = x * t1[nt] + t0[nt];
        float ev = sEb[sl * LP + n];   // residual e from the staged tile
        out[off] = ev + fmaxf(y, 0.0f);
      }
    }
  }
}

extern "C" void kernel_launch(void* const* d_in, const int* in_sizes, int n_in,
                              void* d_out, int out_size, void* d_ws, size_t ws_size,
                              hipStream_t stream) {
  (void)in_sizes; (void)n_in; (void)out_size; (void)ws_size;
  const float* h1 = (const float*)d_in[0];
  const float* h2 = (const float*)d_in[1];
  const float* e = (const float*)d_in[2];
  // d_in[3] = graph (unused, aggregation='sum')
  const float* W_U1 = (const float*)d_in[4];  const float* b_U1 = (const float*)d_in[5];
  const float* W_U2 = (const float*)d_in[6];  const float* b_U2 = (const float*)d_in[7];
  const float* W_V1 = (const float*)d_in[8];  const float* b_V1 = (const float*)d_in[9];
  const float* W_V2 = (const float*)d_in[10]; const float* b_V2 = (const float*)d_in[11];
  const float* W_A  = (const float*)d_in[12]; const float* b_A  = (const float*)d_in[13];
  const float* W_B  = (const float*)d_in[14]; const float* b_B  = (const float*)d_in[15];
  const float* W_C  = (const float*)d_in[16]; const float* b_C  = (const float*)d_in[17];
  const float* gamma_h = (const float*)d_in[18];
  const float* beta_h  = (const float*)d_in[19];
  const float* gamma_e = (const float*)d_in[20];
  const float* beta_e  = (const float*)d_in[21];
  float* ws = (float*)d_ws;
  float* out = (float*)d_out;

  zero_acc<<<(ZERO_COUNT + 255) / 256, 256, 0, stream>>>(ws);

  GemmArgs ga;
  ga.x[0] = h1; ga.x[1] = h2; ga.x[2] = h1; ga.x[3] = h2; ga.x[4] = h1; ga.x[5] = h2;
  ga.W[0] = W_U1; ga.W[1] = W_U2; ga.W[2] = W_V1; ga.W[3] = W_V2; ga.W[4] = W_A; ga.W[5] = W_B;
  ga.bias[0] = b_U1; ga.bias[1] = b_U2; ga.bias[2] = b_V1; ga.bias[3] = b_V2;
  ga.bias[4] = b_A;  ga.bias[5] = b_B;
  ga.out[0] = ws + OFF_UH1; ga.out[1] = ws + OFF_UH2;
  ga.out[2] = ws + OFF_VH1; ga.out[3] = ws + OFF_VH2;
  ga.out[4] = ws + OFF_AH1; ga.out[5] = ws + OFF_BH2;
  small_gemms<<<48, 256, 0, stream>>>(ga);

  pass1<<<1024, 128, 0, stream>>>(e, W_C, b_C, ws);
  stats<<<3, 256, 0, stream>>>(ws);
  apply_h<<<512, 256, 0, stream>>>(h1, h2, gamma_h, beta_h, ws, out);
  pass2<<<1024, 128, 0, stream>>>(e, W_C, b_C, gamma_e, beta_e, ws, out + 131072);
}